// GNN_81681688035648
// MI455X (gfx1250) — compile-verified
//
#include <hip/hip_runtime.h>

typedef float v2f __attribute__((ext_vector_type(2)));
typedef float v8f __attribute__((ext_vector_type(8)));

#define CFEAT 128
#define LDS_STRIDE 132   // 128 + 4 pad; row stride 528B keeps 16B alignment, spreads banks
#define TILES_PER_BLOCK 8   // 8 M-tiles = 128 nodes per block; weights stay in VGPRs

// ---------------- zero fill ----------------
__global__ void zero_f(float* __restrict__ p, size_t n) {
    size_t i = (size_t)blockIdx.x * blockDim.x + threadIdx.x;
    size_t stride = (size_t)gridDim.x * blockDim.x;
    for (; i < n; i += stride) p[i] = 0.0f;
}

// ---------------- edge gather + atomic scatter: one wave32 per edge ----------------
// accumDeg != 0: lane 0 also counts in-degree (layer-1 pass only)
__global__ void __launch_bounds__(256)
scatter_kernel(const float* __restrict__ feat,
               const long long* __restrict__ src,
               const long long* __restrict__ dst,
               float* __restrict__ agg, float* __restrict__ deg,
               int E, int accumDeg) {
    int e = blockIdx.x * (blockDim.x >> 5) + (threadIdx.x >> 5);
    if (e >= E) return;
    int lane = threadIdx.x & 31;
    int s = (int)src[e];
    int d = (int)dst[e];
    const float4 v = *reinterpret_cast<const float4*>(feat + (size_t)s * CFEAT + lane * 4);
    float* ap = agg + (size_t)d * CFEAT + lane * 4;
    atomicAdd(ap + 0, v.x);
    atomicAdd(ap + 1, v.y);
    atomicAdd(ap + 2, v.z);
    atomicAdd(ap + 3, v.w);
    if (accumDeg && lane == 0) atomicAdd(&deg[d], 1.0f);
}

// ---------------- fused SAGE layer: relu((agg/deg)@Wl + x@Wr + b) via f32 WMMA ----------------
// block = 256 threads = 8 waves; wave w owns output cols [16w,16w+16).
// B-fragments of both weight matrices live in registers (v2f[32] each) and are
// reused across TILES_PER_BLOCK M-tiles; only the LHS is restaged in LDS per tile.
__global__ void __launch_bounds__(256)
sage_wmma_kernel(const float* __restrict__ xin,
                 const float* __restrict__ agg,
                 const float* __restrict__ deg,
                 const float* __restrict__ Wl,
                 const float* __restrict__ bias,
                 const float* __restrict__ Wr,
                 float* __restrict__ out, int N) {
    __shared__ float lds_a[16 * LDS_STRIDE];  // mean-aggregated neighbor features
    __shared__ float lds_x[16 * LDS_STRIDE];  // self features

    const int t    = threadIdx.x;
    const int lane = t & 31;
    const int wave = t >> 5;                // 0..7 -> N tile
    const int m    = lane & 15;             // A row / D col within tile
    const int kb   = (lane >> 4) << 1;      // 0 or 2: K sub-offset for A/B frags
    const int n    = wave * 16 + m;         // output column

    // ---- load B fragments for this wave's 16 columns, both weights, all K ----
    v2f bl[32], br[32];
#pragma unroll
    for (int ks = 0; ks < 32; ++ks) {
        int k = ks * 4 + kb;
        bl[ks].x = Wl[(size_t)k * CFEAT + n];
        bl[ks].y = Wl[(size_t)(k + 1) * CFEAT + n];
        br[ks].x = Wr[(size_t)k * CFEAT + n];
        br[ks].y = Wr[(size_t)(k + 1) * CFEAT + n];
    }
    const float bv = bias[n];

    const int block_base = blockIdx.x * (16 * TILES_PER_BLOCK);
    const int srow = t >> 4;            // staging row 0..15
    const int scol = (t & 15) * 8;      // staging col 0,8,...,120
    const int row_hi = (lane >> 4) * 8; // D-row offset for this half-wave

    for (int tile = 0; tile < TILES_PER_BLOCK; ++tile) {
        const int base = block_base + tile * 16;
        if (base >= N) break;  // uniform across block: safe around barriers/WMMA

        // ---- stage 16 rows x 128 cols of scaled-agg and self features ----
        {
            int node = base + srow;
            if (node < N) {
                float inv = 1.0f / fmaxf(deg[node], 1.0f);
                const float4* ap = reinterpret_cast<const float4*>(agg + (size_t)node * CFEAT + scol);
                const float4* xp = reinterpret_cast<const float4*>(xin + (size_t)node * CFEAT + scol);
                float4 a0 = ap[0], a1 = ap[1];
                float4 x0 = xp[0], x1 = xp[1];
                a0.x *= inv; a0.y *= inv; a0.z *= inv; a0.w *= inv;
                a1.x *= inv; a1.y *= inv; a1.z *= inv; a1.w *= inv;
                *reinterpret_cast<float4*>(&lds_a[srow * LDS_STRIDE + scol])     = a0;
                *reinterpret_cast<float4*>(&lds_a[srow * LDS_STRIDE + scol + 4]) = a1;
                *reinterpret_cast<float4*>(&lds_x[srow * LDS_STRIDE + scol])     = x0;
                *reinterpret_cast<float4*>(&lds_x[srow * LDS_STRIDE + scol + 4]) = x1;
            }
        }
        __syncthreads();

        // ---- 64 WMMAs: (agg/deg)@Wl then x@Wr, accumulated in one v8f ----
        v8f acc = {};
        const float* la = &lds_a[m * LDS_STRIDE];
        const float* lx = &lds_x[m * LDS_STRIDE];
#pragma unroll
        for (int ks = 0; ks < 32; ++ks) {
            v2f a;
            a.x = la[ks * 4 + kb];
            a.y = la[ks * 4 + kb + 1];
            acc = __builtin_amdgcn_wmma_f32_16x16x4_f32(false, a, false, bl[ks], (short)0, acc, false, false);
        }
#pragma unroll
        for (int ks = 0; ks < 32; ++ks) {
            v2f a;
            a.x = lx[ks * 4 + kb];
            a.y = lx[ks * 4 + kb + 1];
            acc = __builtin_amdgcn_wmma_f32_16x16x4_f32(false, a, false, br[ks], (short)0, acc, false, false);
        }

        // ---- bias + ReLU + store; D: vgpr r, lane l -> row r + 8*(l>>4), col l&15 ----
#pragma unroll
        for (int r = 0; r < 8; ++r) {
            int node = base + row_hi + r;
            if (node < N) {
                float v = acc[r] + bv;
                out[(size_t)node * CFEAT + n] = fmaxf(v, 0.0f);
            }
        }
        __syncthreads();  // LDS reuse protection before next tile's staging
    }
}

// ---------------- final linear [N,128] @ [128,1] + b ----------------
__global__ void __launch_bounds__(256)
final_kernel(const float* __restrict__ h, const float* __restrict__ wlin,
             const float* __restrict__ blin, float* __restrict__ out, int N) {
    int node = blockIdx.x * 8 + (threadIdx.x >> 5);
    int lane = threadIdx.x & 31;
    if (node >= N) return;
    float s = 0.0f;
#pragma unroll
    for (int j = 0; j < 4; ++j) {
        int c = lane + j * 32;
        s += h[(size_t)node * CFEAT + c] * wlin[c];
    }
#pragma unroll
    for (int off = 16; off > 0; off >>= 1) s += __shfl_down(s, off, 32);
    if (lane == 0) out[node] = s + blin[0];
}

extern "C" void kernel_launch(void* const* d_in, const int* in_sizes, int n_in,
                              void* d_out, int out_size, void* d_ws, size_t ws_size,
                              hipStream_t stream) {
    const float*     x    = (const float*)d_in[0];
    const long long* ei   = (const long long*)d_in[1];
    const float*     W1l  = (const float*)d_in[2];
    const float*     b1   = (const float*)d_in[3];
    const float*     W1r  = (const float*)d_in[4];
    const float*     W2l  = (const float*)d_in[5];
    const float*     b2   = (const float*)d_in[6];
    const float*     W2r  = (const float*)d_in[7];
    const float*     Wlin = (const float*)d_in[8];
    const float*     blin = (const float*)d_in[9];

    const int N = in_sizes[0] / CFEAT;
    const int E = in_sizes[1] / 2;
    const long long* src = ei;
    const long long* dst = ei + E;

    // workspace carve-up (deg + agg + h1 + h2 ~= 154.6 MB)
    char* ws = (char*)d_ws;
    size_t off = 0;
    auto carve = [&](size_t bytes) -> float* {
        off = (off + 255) & ~(size_t)255;
        float* p = (float*)(ws + off);
        off += bytes;
        return p;
    };
    float* deg = carve((size_t)N * sizeof(float));
    float* agg = carve((size_t)N * CFEAT * sizeof(float));
    float* h1  = carve((size_t)N * CFEAT * sizeof(float));
    float* h2  = carve((size_t)N * CFEAT * sizeof(float));
    (void)ws_size; (void)n_in; (void)out_size;

    const dim3 blk(256);
    const dim3 zgrid(2048);
    const dim3 egrid8((E + 7) / 8);
    const dim3 ngridS((N + 16 * TILES_PER_BLOCK - 1) / (16 * TILES_PER_BLOCK));
    const dim3 ngrid8((N + 7) / 8);

    // layer 1 (degree fused into first scatter)
    hipLaunchKernelGGL(zero_f, zgrid, blk, 0, stream, deg, (size_t)N);
    hipLaunchKernelGGL(zero_f, zgrid, blk, 0, stream, agg, (size_t)N * CFEAT);
    hipLaunchKernelGGL(scatter_kernel, egrid8, blk, 0, stream, x, src, dst, agg, deg, E, 1);
    hipLaunchKernelGGL(sage_wmma_kernel, ngridS, blk, 0, stream, x, agg, deg, W1l, b1, W1r, h1, N);
    // layer 2
    hipLaunchKernelGGL(zero_f, zgrid, blk, 0, stream, agg, (size_t)N * CFEAT);
    hipLaunchKernelGGL(scatter_kernel, egrid8, blk, 0, stream, h1, src, dst, agg, deg, E, 0);
    hipLaunchKernelGGL(sage_wmma_kernel, ngridS, blk, 0, stream, h1, agg, deg, W2l, b2, W2r, h2, N);
    // readout
    hipLaunchKernelGGL(final_kernel, ngrid8, blk, 0, stream, h2, Wlin, blin, (float*)d_out, N);
}